// KiperwasserDependencyParser_62388694941738
// MI455X (gfx1250) — compile-verified
//
#include <hip/hip_runtime.h>
#include <hip/hip_bf16.h>
#include <cstddef>

// ---------------------------------------------------------------------------
// Kiperwasser dependency parser for MI455X (gfx1250, wave32, WMMA).
//   V=512 T=64 H=256 N=1024 HID=100 D=576
// GEMMs use native f32 WMMA (V_WMMA_F32_16X16X4_F32): GEMM flops (~5 GF) are
// negligible vs the serialized LSTM scan, so we keep full f32 precision.
// ---------------------------------------------------------------------------

typedef float v2f __attribute__((ext_vector_type(2)));
typedef float v8f __attribute__((ext_vector_type(8)));

#define N_SEQ 1024
#define HDIM  256
#define DDIM  576
#define G4H   1024   // 4*H
#define HID_DIM 100

__device__ __forceinline__ float fast_sigmoid(float x) {
    return 1.0f / (1.0f + __expf(-x));
}
__device__ __forceinline__ float fast_tanh(float x) {
    // tanh(x) = 2*sigmoid(2x) - 1  (single TRANS op path)
    return 2.0f * fast_sigmoid(2.0f * x) - 1.0f;
}
__device__ __forceinline__ v2f ldv2(const float* p) {
    return *(const v2f*)p;   // 8B-aligned by construction -> global_load_b64
}
#define KDP_WMMA_F32(A, B, ACC) \
    __builtin_amdgcn_wmma_f32_16x16x4_f32(false, (A), false, (B), (short)0, (ACC), false, false)

// ---------------------------------------------------------------------------
// 1) Embedding gather: x[i, 0:512] = word_table[widx[i]], x[i, 512:576] = pos.
// ---------------------------------------------------------------------------
__global__ void kdp_embed(const float* __restrict__ wt, const float* __restrict__ pt,
                          const int* __restrict__ widx, const int* __restrict__ pidx,
                          float* __restrict__ x) {
    int i = blockIdx.x;                 // sequence position 0..1023
    int wi = widx[i], pi = pidx[i];
    for (int c = threadIdx.x; c < DDIM; c += blockDim.x) {
        float v = (c < 512) ? wt[wi * 512 + c] : pt[pi * 64 + (c - 512)];
        x[(size_t)i * DDIM + c] = v;
    }
}

// ---------------------------------------------------------------------------
// 2) f32 WMMA GEMM:  C[M x Nout] = X[M x K] @ W[Nout x K]^T  (+ bias[Nout])
//    Each wave computes a 16x64 C strip: one A fragment is reused across 4
//    B fragments -> 4x V_WMMA_F32_16X16X4_F32 per 5 b64 loads per K-step.
//    Software-pipelined: next K-step's fragments are loaded before this
//    step's WMMAs so load latency overlaps matrix math. B row indices are
//    CLAMPED (not predicated) so the inner loop has no EXEC manipulation;
//    out-of-range columns are masked only at the store.
//    A-frag layout (ISA 7.12.2):  a[v] = A[m = lane&15][k0 + 2*(lane>>4) + v]
//    C/D layout: vgpr r, lane L -> C[r + 8*(L>>4)][L&15]
// ---------------------------------------------------------------------------
__global__ __launch_bounds__(128) void kdp_wmma_gemm(
        const float* __restrict__ X, int ldx,
        const float* __restrict__ W, int ldw,
        const float* __restrict__ bias,
        float* __restrict__ C, int ldc,
        int M, int K, int Nout) {
    int wave = threadIdx.x >> 5;
    int lane = threadIdx.x & 31;
    int mbase = blockIdx.y * 16;
    int nbase = (blockIdx.x * 4 + wave) * 64;    // wave-uniform
    if (nbase >= Nout) return;                   // EXEC stays all-1s

    int lo = lane & 15;
    int hi = lane >> 4;
    const float* arow = X + (size_t)(mbase + lo) * ldx;

    int r0 = nbase + lo;
    int r1 = r0 + 16, r2 = r0 + 32, r3 = r0 + 48;
    int cl = Nout - 1;
    if (r0 > cl) r0 = cl;
    if (r1 > cl) r1 = cl;
    if (r2 > cl) r2 = cl;
    if (r3 > cl) r3 = cl;
    const float* b0p = W + (size_t)r0 * ldw;
    const float* b1p = W + (size_t)r1 * ldw;
    const float* b2p = W + (size_t)r2 * ldw;
    const float* b3p = W + (size_t)r3 * ldw;

    v8f acc0 = {}, acc1 = {}, acc2 = {}, acc3 = {};

    // --- software-pipelined K loop: prologue loads step 0 ---
    int kk = 2 * hi;
    v2f a  = ldv2(arow + kk);
    v2f b0 = ldv2(b0p + kk);
    v2f b1 = ldv2(b1p + kk);
    v2f b2 = ldv2(b2p + kk);
    v2f b3 = ldv2(b3p + kk);
#pragma unroll 2
    for (int k0 = 4; k0 < K; k0 += 4) {
        int kn = k0 + 2 * hi;
        // prefetch next step while this step's WMMAs run
        v2f an  = ldv2(arow + kn);
        v2f bn0 = ldv2(b0p + kn);
        v2f bn1 = ldv2(b1p + kn);
        v2f bn2 = ldv2(b2p + kn);
        v2f bn3 = ldv2(b3p + kn);
        acc0 = KDP_WMMA_F32(a, b0, acc0);
        acc1 = KDP_WMMA_F32(a, b1, acc1);
        acc2 = KDP_WMMA_F32(a, b2, acc2);
        acc3 = KDP_WMMA_F32(a, b3, acc3);
        a = an; b0 = bn0; b1 = bn1; b2 = bn2; b3 = bn3;
    }
    acc0 = KDP_WMMA_F32(a, b0, acc0);
    acc1 = KDP_WMMA_F32(a, b1, acc1);
    acc2 = KDP_WMMA_F32(a, b2, acc2);
    acc3 = KDP_WMMA_F32(a, b3, acc3);

#define KDP_STORE_TILE(i, ACC)                                              \
    {                                                                       \
        int ncol = nbase + 16 * (i) + lo;                                   \
        if (ncol < Nout) {                                                  \
            float bv = bias ? bias[ncol] : 0.0f;                            \
            _Pragma("unroll")                                               \
            for (int r = 0; r < 8; ++r) {                                   \
                int m = mbase + r + 8 * hi;                                 \
                C[(size_t)m * ldc + ncol] = (ACC)[r] + bv;                  \
            }                                                               \
        }                                                                   \
    }
    KDP_STORE_TILE(0, acc0)
    KDP_STORE_TILE(1, acc1)
    KDP_STORE_TILE(2, acc2)
    KDP_STORE_TILE(3, acc3)
#undef KDP_STORE_TILE
}

// ---------------------------------------------------------------------------
// 3) Transpose Whh (1024x256) -> WhhT (256x1024) so the recurrent matvec
//    streams coalesced (lane = output column). 4 matrices, blockIdx.z picks.
// ---------------------------------------------------------------------------
__global__ void kdp_transpose4(const float* __restrict__ A0, const float* __restrict__ A1,
                               const float* __restrict__ A2, const float* __restrict__ A3,
                               float* __restrict__ T, int rows, int cols) {
    const float* A = (blockIdx.z == 0) ? A0 : (blockIdx.z == 1) ? A1
                   : (blockIdx.z == 2) ? A2 : A3;
    float* Tz = T + (size_t)blockIdx.z * rows * cols;
    __shared__ float tile[32][33];
    int r0 = blockIdx.y * 32, c0 = blockIdx.x * 32;
    int tx = threadIdx.x, ty = threadIdx.y;     // 32 x 8
#pragma unroll
    for (int i = 0; i < 32; i += 8)
        tile[ty + i][tx] = A[(size_t)(r0 + ty + i) * cols + (c0 + tx)];
    __syncthreads();
#pragma unroll
    for (int i = 0; i < 32; i += 8)
        Tz[(size_t)(c0 + ty + i) * rows + (r0 + tx)] = tile[tx][ty + i];
}

// ---------------------------------------------------------------------------
// 4) Sequential LSTM scan. One block per direction (fwd/bwd run concurrently
//    on separate WGPs). 256 threads; thread j owns gate rows 4j..4j+3 for the
//    matvec (float4 b128 loads of WhhT -> 4 FMAs per VMEM instruction) and
//    owns cell j for the elementwise update (c held in a register).
// ---------------------------------------------------------------------------
__global__ __launch_bounds__(256, 1) void kdp_lstm_scan(
        const float* __restrict__ xw_fwd, const float* __restrict__ whhT_fwd,
        const float* __restrict__ xw_bwd, const float* __restrict__ whhT_bwd,
        float* __restrict__ hout /* N x 512 */, int nT) {
    bool rev = (blockIdx.x == 1);
    const float* xw = rev ? xw_bwd : xw_fwd;
    const float4* WT4 = (const float4*)(rev ? whhT_bwd : whhT_fwd); // 256 x (1024/4)
    int colOff = rev ? HDIM : 0;

    __shared__ float hsm[HDIM];
    __shared__ float gates[G4H];

    int j = threadIdx.x;          // 0..255
    hsm[j] = 0.0f;
    float c = 0.0f;
    __syncthreads();

    for (int s = 0; s < nT; ++s) {
        int t = rev ? (nT - 1 - s) : s;
        // --- gate matvec: acc = xw[t][4j..4j+3] + WhhT^T h ---
        float4 acc = ((const float4*)(xw + (size_t)t * G4H))[j];
#pragma unroll 8
        for (int k = 0; k < HDIM; ++k) {
            float4 w = WT4[(size_t)k * (G4H / 4) + j];
            float h = hsm[k];
            acc.x = fmaf(w.x, h, acc.x);
            acc.y = fmaf(w.y, h, acc.y);
            acc.z = fmaf(w.z, h, acc.z);
            acc.w = fmaf(w.w, h, acc.w);
        }
        ((float4*)gates)[j] = acc;      // ds_store_b128
        __syncthreads();                // gates ready, all hsm reads done
        // --- cell update (thread j owns hidden unit j) ---
        float ig = fast_sigmoid(gates[j]);
        float fg = fast_sigmoid(gates[HDIM + j]);
        float gg = fast_tanh   (gates[2 * HDIM + j]);
        float og = fast_sigmoid(gates[3 * HDIM + j]);
        c = fg * c + ig * gg;
        float h = og * fast_tanh(c);
        hsm[j] = h;
        hout[(size_t)t * (2 * HDIM) + colOff + j] = h;
        __syncthreads();                // hsm ready for next step
    }
}

// ---------------------------------------------------------------------------
// 5) Fused pairwise scorer: scores[i,j] = W2 . tanh(A[i]+B[j]+b1) + b2,
//    diagonal zeroed. 16x16 tile per block; A/B/W2 staged in LDS; the
//    (N,N,HID) intermediate is never materialized.
// ---------------------------------------------------------------------------
__global__ __launch_bounds__(256) void kdp_score(
        const float* __restrict__ A, const float* __restrict__ B,
        const float* __restrict__ b1, const float* __restrict__ W2,
        const float* __restrict__ b2, float* __restrict__ out, int Nn) {
    __shared__ float As[16][101];
    __shared__ float Bs[16][101];
    __shared__ float w2s[HID_DIM];
    int itile = blockIdx.y * 16, jtile = blockIdx.x * 16;
    int t = threadIdx.x;
    for (int idx = t; idx < 16 * HID_DIM; idx += 256) {
        int r = idx / HID_DIM, k = idx - r * HID_DIM;
        As[r][k] = A[(size_t)(itile + r) * HID_DIM + k] + b1[k];
        Bs[r][k] = B[(size_t)(jtile + r) * HID_DIM + k];
    }
    if (t < HID_DIM) w2s[t] = W2[t];
    __syncthreads();

    int ti = t >> 4, tj = t & 15;
    float s = b2[0];
#pragma unroll 4
    for (int k = 0; k < HID_DIM; ++k)
        s = fmaf(w2s[k], fast_tanh(As[ti][k] + Bs[tj][k]), s);
    int gi = itile + ti, gj = jtile + tj;
    out[(size_t)gi * Nn + gj] = (gi == gj) ? 0.0f : s;
}

// ---------------------------------------------------------------------------
// 6) In-place softmax over axis 0 (down each column). One block per column;
//    each thread keeps its 4 values in registers (single global read).
// ---------------------------------------------------------------------------
__global__ __launch_bounds__(256) void kdp_softmax_col(float* __restrict__ out, int Nn) {
    int j = blockIdx.x;
    int t = threadIdx.x;
    __shared__ float red[256];
    float vals[4];
    float mx = -3.0e38f;
#pragma unroll
    for (int r = 0; r < 4; ++r) {
        vals[r] = out[(size_t)(t + 256 * r) * Nn + j];
        mx = fmaxf(mx, vals[r]);
    }
    red[t] = mx; __syncthreads();
    for (int s = 128; s > 0; s >>= 1) {
        if (t < s) red[t] = fmaxf(red[t], red[t + s]);
        __syncthreads();
    }
    mx = red[0]; __syncthreads();
    float sum = 0.0f;
#pragma unroll
    for (int r = 0; r < 4; ++r) { vals[r] = __expf(vals[r] - mx); sum += vals[r]; }
    red[t] = sum; __syncthreads();
    for (int s = 128; s > 0; s >>= 1) {
        if (t < s) red[t] += red[t + s];
        __syncthreads();
    }
    float inv = 1.0f / red[0];
#pragma unroll
    for (int r = 0; r < 4; ++r)
        out[(size_t)(t + 256 * r) * Nn + j] = vals[r] * inv;
}

// ---------------------------------------------------------------------------
// Host-side orchestration.
// ---------------------------------------------------------------------------
extern "C" void kernel_launch(void* const* d_in, const int* in_sizes, int n_in,
                              void* d_out, int out_size, void* d_ws, size_t ws_size,
                              hipStream_t stream) {
    const float* word_table = (const float*)d_in[0];
    const float* pos_table  = (const float*)d_in[1];
    const float* l0f_Wih = (const float*)d_in[2];
    const float* l0f_Whh = (const float*)d_in[3];
    const float* l0f_b   = (const float*)d_in[4];
    const float* l0b_Wih = (const float*)d_in[5];
    const float* l0b_Whh = (const float*)d_in[6];
    const float* l0b_b   = (const float*)d_in[7];
    const float* l1f_Wih = (const float*)d_in[8];
    const float* l1f_Whh = (const float*)d_in[9];
    const float* l1f_b   = (const float*)d_in[10];
    const float* l1b_Wih = (const float*)d_in[11];
    const float* l1b_Whh = (const float*)d_in[12];
    const float* l1b_b   = (const float*)d_in[13];
    const float* mlp_W1  = (const float*)d_in[14];   // (100, 1024)
    const float* mlp_b1  = (const float*)d_in[15];
    const float* mlp_W2  = (const float*)d_in[16];   // (1, 100)
    const float* mlp_b2  = (const float*)d_in[17];
    const int*   word_idx = (const int*)d_in[18];
    const int*   pos_idx  = (const int*)d_in[19];
    float* out = (float*)d_out;                      // (1024, 1024)

    // Workspace carve-up (floats).
    float* ws = (float*)d_ws;
    size_t off = 0;
    float* x    = ws + off; off += (size_t)N_SEQ * DDIM;        // 1024x576
    float* xw_f = ws + off; off += (size_t)N_SEQ * G4H;         // 1024x1024
    float* xw_b = ws + off; off += (size_t)N_SEQ * G4H;
    float* h0   = ws + off; off += (size_t)N_SEQ * 2 * HDIM;    // 1024x512
    float* h1   = ws + off; off += (size_t)N_SEQ * 2 * HDIM;
    float* Abuf = ws + off; off += (size_t)N_SEQ * HID_DIM;     // 1024x100
    float* Bbuf = ws + off; off += (size_t)N_SEQ * HID_DIM;
    float* whhT = ws + off; off += (size_t)4 * HDIM * G4H;      // 4 x 256x1024

    // 0) transpose all four Whh matrices (independent of everything else)
    kdp_transpose4<<<dim3(HDIM / 32, G4H / 32, 4), dim3(32, 8), 0, stream>>>(
        l0f_Whh, l0b_Whh, l1f_Whh, l1b_Whh, whhT, G4H, HDIM);

    // 1) embeddings
    kdp_embed<<<N_SEQ, 128, 0, stream>>>(word_table, pos_table, word_idx, pos_idx, x);

    // 2) layer-0 input GEMMs: xw = x @ Wih^T + b   (M=1024, K=576, N=1024)
    //    each wave covers 64 cols -> 16 waves across N, 4 waves/block
    dim3 gemmGrid(G4H / 64 / 4, N_SEQ / 16);
    kdp_wmma_gemm<<<gemmGrid, 128, 0, stream>>>(x, DDIM, l0f_Wih, DDIM, l0f_b,
                                                xw_f, G4H, N_SEQ, DDIM, G4H);
    kdp_wmma_gemm<<<gemmGrid, 128, 0, stream>>>(x, DDIM, l0b_Wih, DDIM, l0b_b,
                                                xw_b, G4H, N_SEQ, DDIM, G4H);

    // 3) layer-0 scan (fwd + bwd concurrently) -> h0 (concat into columns)
    kdp_lstm_scan<<<2, 256, 0, stream>>>(xw_f, whhT + 0 * (size_t)HDIM * G4H,
                                         xw_b, whhT + 1 * (size_t)HDIM * G4H,
                                         h0, N_SEQ);

    // 4) layer-1 input GEMMs: K = 512
    kdp_wmma_gemm<<<gemmGrid, 128, 0, stream>>>(h0, 2 * HDIM, l1f_Wih, 2 * HDIM, l1f_b,
                                                xw_f, G4H, N_SEQ, 2 * HDIM, G4H);
    kdp_wmma_gemm<<<gemmGrid, 128, 0, stream>>>(h0, 2 * HDIM, l1b_Wih, 2 * HDIM, l1b_b,
                                                xw_b, G4H, N_SEQ, 2 * HDIM, G4H);

    // 5) layer-1 scan -> h1
    kdp_lstm_scan<<<2, 256, 0, stream>>>(xw_f, whhT + 2 * (size_t)HDIM * G4H,
                                         xw_b, whhT + 3 * (size_t)HDIM * G4H,
                                         h1, N_SEQ);

    // 6) A = h1 @ Wh^T, B = h1 @ Wm^T  (Wh = W1[:, :512], Wm = W1[:, 512:])
    //    Nout=100 -> 2 active waves (64-col strips), 1 block across N
    dim3 abGrid(1, N_SEQ / 16);
    kdp_wmma_gemm<<<abGrid, 128, 0, stream>>>(h1, 2 * HDIM, mlp_W1, G4H, nullptr,
                                              Abuf, HID_DIM, N_SEQ, 2 * HDIM, HID_DIM);
    kdp_wmma_gemm<<<abGrid, 128, 0, stream>>>(h1, 2 * HDIM, mlp_W1 + 2 * HDIM, G4H, nullptr,
                                              Bbuf, HID_DIM, N_SEQ, 2 * HDIM, HID_DIM);

    // 7) fused pairwise scores into d_out (diagonal zeroed)
    kdp_score<<<dim3(N_SEQ / 16, N_SEQ / 16), 256, 0, stream>>>(
        Abuf, Bbuf, mlp_b1, mlp_W2, mlp_b2, out, N_SEQ);

    // 8) softmax down each column, in place
    kdp_softmax_col<<<N_SEQ, 256, 0, stream>>>(out, N_SEQ);
}